// MACE_76613626626643
// MI455X (gfx1250) — compile-verified
//
#include <hip/hip_runtime.h>

// ---------------- problem constants (match reference) ----------------
#define NN 10000      // nodes
#define NE 320000     // edges
#define C  64         // channels
#define L  16         // spherical components
#define NB 8          // bessel
#define NS 5          // species
#define O  64         // out per layer
#define T  2          // layers
#define NTILE (NN/16) // 625 node tiles of 16

typedef __attribute__((ext_vector_type(2))) float v2f;
typedef __attribute__((ext_vector_type(8))) float v8f;

__device__ __forceinline__ v8f v8zero() {
  v8f v = {0.f,0.f,0.f,0.f,0.f,0.f,0.f,0.f};
  return v;
}

// D = A(16x4 f32) * B(4x16 f32) + C ; full-precision f32 WMMA
__device__ __forceinline__ v8f wmma4(v2f a, v2f b, v8f c) {
  return __builtin_amdgcn_wmma_f32_16x16x4_f32(
      /*neg_a=*/false, a, /*neg_b=*/false, b,
      /*c_mod=*/(short)0, c, /*reuse_a=*/false, /*reuse_b=*/false);
}

// ---------------- init: h[n, l=0, c] = embed[species[n], c] ----------------
__global__ void __launch_bounds__(256) mace_init_h(
    const int* __restrict__ species, const float* __restrict__ embed,
    float* __restrict__ h)
{
  int idx = blockIdx.x * blockDim.x + threadIdx.x;
  if (idx >= NN * C) return;
  int n = idx >> 6;
  int c = idx & 63;
  h[(size_t)n * (C * L) + c] = embed[species[n] * C + c];
}

// ---------------- edge kernel: one wave per edge ----------------
// h   : (N, L, C)  -> sender block is contiguous 4KB
// agg : (L, C, N)  -> coalesced WMMA A-loads later
__global__ void __launch_bounds__(256) mace_edge(
    const float* __restrict__ pos, const int* __restrict__ senders,
    const int* __restrict__ receivers, const float* __restrict__ w_radial_t,
    const float* __restrict__ h, float* __restrict__ agg)
{
  const int lane = threadIdx.x & 31;
  const int e = blockIdx.x * 8 + (threadIdx.x >> 5);
  if (e >= NE) return;

  const int s = senders[e];
  const int r = receivers[e];

  const float rx = pos[r * 3 + 0] - pos[s * 3 + 0];
  const float ry = pos[r * 3 + 1] - pos[s * 3 + 1];
  const float rz = pos[r * 3 + 2] - pos[s * 3 + 2];
  const float d  = sqrtf(rx * rx + ry * ry + rz * rz);

  // envelope: exactly zero for d >= RMAX  -> whole edge contributes nothing
  const float xe = 2.0f - 0.4f * d;            // 2*(1 - d/5)
  if (xe <= 0.0f) return;                      // wave-uniform early out
  const float env = 1.9784613f * expf(-1.0f / xe);  // 1.2*exp(0.5)*exp(-1/x)

  const float ds  = fmaxf(d, 1e-9f);
  const float inv = 1.0f / ds;
  const float x = rx * inv, y = ry * inv, z = rz * inv;
  const float x2 = x * x, y2 = y * y, z2 = z * z;

  // real spherical harmonics (l<=3), exact constants from reference
  float Y[16];
  Y[0]  = 1.0f;
  Y[1]  = 1.7320508f * x;
  Y[2]  = 1.7320508f * y;
  Y[3]  = 1.7320508f * z;
  Y[4]  = 3.8729833f * x * y;
  Y[5]  = 3.8729833f * y * z;
  Y[6]  = 1.1180340f * (3.0f * z2 - 1.0f);
  Y[7]  = 3.8729833f * x * z;
  Y[8]  = 1.9364917f * (x2 - y2);
  Y[9]  = 2.0916501f * y * (3.0f * x2 - y2);
  Y[10] = 10.246951f * x * y * z;
  Y[11] = 1.6201852f * y * (5.0f * z2 - 1.0f);
  Y[12] = 1.3228757f * z * (5.0f * z2 - 3.0f);
  Y[13] = 1.6201852f * x * (5.0f * z2 - 1.0f);
  Y[14] = 5.1234754f * z * (x2 - y2);
  Y[15] = 2.0916501f * x * (x2 - 3.0f * y2);

  // bessel basis via sin recurrence: sin((n+1)*theta)
  const float theta = 0.62831853f * ds;        // pi*d_safe/RMAX
  float rb[NB];
  {
    const float sn = sinf(theta), cn = cosf(theta);
    const float coef = 0.63245553f * inv;      // sqrt(2/RMAX)/d_safe
    const float twoc = 2.0f * cn;
    float sp = 0.0f, sc = sn;
    #pragma unroll
    for (int n = 0; n < NB; ++n) {
      rb[n] = coef * sc;
      float nx = twoc * sc - sp;
      sp = sc; sc = nx;
    }
  }

  // radial MLP: rad = (rb @ w_radial) * env ; lane owns channels c0, c0+1
  const int c0 = lane * 2;
  float w1a = 0.f, w1b = 0.f, w2a = 0.f, w2b = 0.f;
  #pragma unroll
  for (int n = 0; n < NB; ++n) {
    const float* wr = w_radial_t + n * (2 * C);
    w1a += rb[n] * wr[c0];
    w1b += rb[n] * wr[c0 + 1];
    w2a += rb[n] * wr[C + c0];
    w2b += rb[n] * wr[C + c0 + 1];
  }
  w1a *= env; w1b *= env; w2a *= env; w2b *= env;

  const float* hs = h + (size_t)s * (C * L);
  const float hs0a = hs[c0];
  const float hs0b = hs[c0 + 1];

  // msg[c,l] = w1[c]*Y[l]*h[s,0,c] + w2[c]*h[s,l,c]  -> atomic into agg[l,c,r]
  #pragma unroll
  for (int l = 0; l < L; ++l) {
    const float2 hv = *(const float2*)(hs + l * C + c0);
    const float yl = Y[l];
    float m0 = w1a * yl * hs0a + w2a * hv.x;
    float m1 = w1b * yl * hs0b + w2b * hv.y;
    unsafeAtomicAdd(agg + (size_t)l * (C * NN) + (size_t)c0       * NN + r, m0);
    unsafeAtomicAdd(agg + (size_t)l * (C * NN) + (size_t)(c0 + 1) * NN + r, m1);
  }
}

// ---------------- node kernel: one wave per 16-node tile, f32 WMMA ----------------
// agg : (L, C, N), h out : (N, L, C)
__global__ void __launch_bounds__(128) mace_node(
    const float* __restrict__ agg, const int* __restrict__ species,
    const float* __restrict__ w_lin_t,   // (4, C, C)
    const float* __restrict__ w_self_t,  // (NS, C)
    float* __restrict__ h)
{
  const int lane = threadIdx.x & 31;
  const int tile = blockIdx.x * 4 + (threadIdx.x >> 5);
  if (tile >= NTILE) return;
  const int nb    = tile * 16;
  const int row   = lane & 15;          // M (node) for A, N (d) for B
  const int kh    = (lane >> 4) * 2;    // K pair select
  const int nhalf = (lane >> 4) * 8;    // node offset for C/D layout
  const float invAVG = 1.0f / 32.0f;

  v8f gate[4];

  // ---- l = 0 : hn0 = (agg0/AVG) @ W0 + w_self ; gate = sigmoid ; h0 = silu ----
  {
    v8f acc[4] = {v8zero(), v8zero(), v8zero(), v8zero()};
    const float* W = w_lin_t;  // L_OF[0] == 0
    #pragma unroll
    for (int kb = 0; kb < C; kb += 4) {
      const float* ap = agg + (size_t)(kb + kh) * NN + nb + row;
      v2f a; a.x = ap[0] * invAVG; a.y = ap[NN] * invAVG;
      #pragma unroll
      for (int dt = 0; dt < 4; ++dt) {
        v2f b;
        b.x = W[(kb + kh)     * C + dt * 16 + row];
        b.y = W[(kb + kh + 1) * C + dt * 16 + row];
        acc[dt] = wmma4(a, b, acc[dt]);
      }
    }
    #pragma unroll
    for (int dt = 0; dt < 4; ++dt) {
      #pragma unroll
      for (int i = 0; i < 8; ++i) {
        const int node = nb + nhalf + i;
        const int dcol = dt * 16 + row;
        float sv = acc[dt][i] + w_self_t[species[node] * C + dcol];
        float sg = 1.0f / (1.0f + expf(-sv));
        gate[dt][i] = sg;
        h[(size_t)node * (C * L) + dcol] = sv * sg;   // l=0 slice: silu
      }
    }
  }

  // ---- l = 1..15 : h_l = ((agg_l/AVG) @ W[L_OF[l]]) * gate ----
  for (int l = 1; l < L; ++l) {
    const int g = (l >= 9) ? 3 : ((l >= 4) ? 2 : 1);   // L_OF[l]
    const float* W    = w_lin_t + (size_t)g * C * C;
    const float* aggl = agg + (size_t)l * (C * NN);
    v8f acc[4] = {v8zero(), v8zero(), v8zero(), v8zero()};
    #pragma unroll
    for (int kb = 0; kb < C; kb += 4) {
      const float* ap = aggl + (size_t)(kb + kh) * NN + nb + row;
      v2f a; a.x = ap[0] * invAVG; a.y = ap[NN] * invAVG;
      #pragma unroll
      for (int dt = 0; dt < 4; ++dt) {
        v2f b;
        b.x = W[(kb + kh)     * C + dt * 16 + row];
        b.y = W[(kb + kh + 1) * C + dt * 16 + row];
        acc[dt] = wmma4(a, b, acc[dt]);
      }
    }
    #pragma unroll
    for (int dt = 0; dt < 4; ++dt) {
      #pragma unroll
      for (int i = 0; i < 8; ++i) {
        const int node = nb + nhalf + i;
        h[(size_t)node * (C * L) + l * C + dt * 16 + row] = acc[dt][i] * gate[dt][i];
      }
    }
  }
}

// ---------------- output projection: out[:, toff:toff+64] = h0 @ w_out_t ----------------
__global__ void __launch_bounds__(128) mace_out(
    const float* __restrict__ h, const float* __restrict__ w_out_t,
    float* __restrict__ out, int toff)
{
  const int lane = threadIdx.x & 31;
  const int tile = blockIdx.x * 4 + (threadIdx.x >> 5);
  if (tile >= NTILE) return;
  const int nb    = tile * 16;
  const int row   = lane & 15;
  const int kh    = (lane >> 4) * 2;
  const int nhalf = (lane >> 4) * 8;

  v8f acc[4] = {v8zero(), v8zero(), v8zero(), v8zero()};
  #pragma unroll
  for (int kb = 0; kb < C; kb += 4) {
    const float* ap = h + (size_t)(nb + row) * (C * L) + kb + kh; // l=0 slice
    v2f a; a.x = ap[0]; a.y = ap[1];
    #pragma unroll
    for (int dt = 0; dt < 4; ++dt) {
      v2f b;
      b.x = w_out_t[(kb + kh)     * O + dt * 16 + row];
      b.y = w_out_t[(kb + kh + 1) * O + dt * 16 + row];
      acc[dt] = wmma4(a, b, acc[dt]);
    }
  }
  #pragma unroll
  for (int dt = 0; dt < 4; ++dt) {
    #pragma unroll
    for (int i = 0; i < 8; ++i) {
      out[(size_t)(nb + nhalf + i) * (2 * O) + toff + dt * 16 + row] = acc[dt][i];
    }
  }
}

// ---------------- host: launch sequence (graph-capture safe) ----------------
extern "C" void kernel_launch(void* const* d_in, const int* in_sizes, int n_in,
                              void* d_out, int out_size, void* d_ws, size_t ws_size,
                              hipStream_t stream) {
  const float* pos       = (const float*)d_in[0];
  const int*   species   = (const int*)  d_in[1];
  const int*   senders   = (const int*)  d_in[2];
  const int*   receivers = (const int*)  d_in[3];
  const float* embed     = (const float*)d_in[4];
  const float* w_radial  = (const float*)d_in[5];
  const float* w_lin     = (const float*)d_in[6];
  const float* w_self    = (const float*)d_in[7];
  const float* w_out     = (const float*)d_in[8];
  float* out = (float*)d_out;

  const size_t hsz = (size_t)NN * C * L;       // elements (10.24M) = 40.96 MB
  float* h   = (float*)d_ws;                   // (N, L, C)
  float* agg = h + hsz;                        // (L, C, N)

  hipMemsetAsync(h, 0, hsz * sizeof(float), stream);
  mace_init_h<<<(NN * C + 255) / 256, 256, 0, stream>>>(species, embed, h);

  for (int t = 0; t < T; ++t) {
    hipMemsetAsync(agg, 0, hsz * sizeof(float), stream);
    mace_edge<<<(NE + 7) / 8, 256, 0, stream>>>(
        pos, senders, receivers, w_radial + (size_t)t * NB * 2 * C, h, agg);
    mace_node<<<(NTILE + 3) / 4, 128, 0, stream>>>(
        agg, species, w_lin + (size_t)t * 4 * C * C, w_self + (size_t)t * NS * C, h);
    mace_out<<<(NTILE + 3) / 4, 128, 0, stream>>>(
        h, w_out + (size_t)t * C * O, out, t * O);
  }
}